// GCN_3075196584115
// MI455X (gfx1250) — compile-verified
//
#include <hip/hip_runtime.h>
#include <hip/hip_bf16.h>

typedef __attribute__((ext_vector_type(2))) float v2f;
typedef __attribute__((ext_vector_type(8))) float v8f;

// ---------------------------------------------------------------------------
// Degree / normalization
// ---------------------------------------------------------------------------
__global__ void deg_init_kernel(float* __restrict__ deg, int n) {
    int i = blockIdx.x * blockDim.x + threadIdx.x;
    if (i < n) deg[i] = 1.0f;  // self loop contributes 1 to every node
}

__global__ void deg_edges_kernel(const long long* __restrict__ dst, int E,
                                 float* __restrict__ deg) {
    int e = blockIdx.x * blockDim.x + threadIdx.x;
    if (e < E) atomicAdd(&deg[(int)dst[e]], 1.0f);
}

__global__ void dinv_kernel(float* __restrict__ deg, int n) {
    int i = blockIdx.x * blockDim.x + threadIdx.x;
    if (i < n) {
        float d = deg[i];
        deg[i] = (d > 0.0f) ? rsqrtf(d) : 0.0f;  // in-place -> dinv
    }
}

// ---------------------------------------------------------------------------
// GEMM: C[M,NCOLS] = A[M,K] * B[K,NCOLS]  via V_WMMA_F32_16X16X4_F32
// One wave computes one 16x16 tile of C. blockDim.x = 256 (8 waves).
//
// Per ISA 7.12.2 (wave32):
//   A 16x4:  lanes 0-15 -> rows M=0..15 with K={0,1} in v[0],v[1];
//            lanes 16-31 -> same rows with K={2,3}
//   B 4x16:  lane holds column N=lane&15; lanes 0-15 -> K={0,1},
//            lanes 16-31 -> K={2,3}
//   C 16x16: lane&15 -> col; VGPR v -> row v + 8*(lane>>4)
// ---------------------------------------------------------------------------
template <int K, int NCOLS>
__global__ void gemm_wmma_f32_kernel(const float* __restrict__ A,
                                     const float* __restrict__ B,
                                     float* __restrict__ C, int M) {
    const int wavesPerBlock = blockDim.x >> 5;
    const int wave = blockIdx.x * wavesPerBlock + (threadIdx.x >> 5);
    const int ntilesN = NCOLS / 16;
    const int tm = wave / ntilesN;
    const int tn = wave % ntilesN;
    const int m0 = tm * 16;
    if (m0 >= M) return;
    const int n0 = tn * 16;

    const int lane = threadIdx.x & 31;
    const int half = lane >> 4;   // 0 or 1
    const int r    = lane & 15;

    // Row for A loads (clamp so edge tiles read in-bounds; stores are guarded)
    int arow = m0 + r;
    if (arow >= M) arow = M - 1;

    const float* __restrict__ aptr = A + (long)arow * K + 2 * half;       // K pair base
    const float* __restrict__ bptr = B + (long)(2 * half) * NCOLS + n0 + r;

    v8f acc = {};
#pragma unroll 8
    for (int kk = 0; kk < K; kk += 4) {
        v2f a = *(const v2f*)(aptr + kk);              // A[row][kk+2h], A[row][kk+2h+1]
        v2f b;
        b.x = bptr[(long)kk * NCOLS];                  // B[kk+2h  ][n0+r]
        b.y = bptr[(long)(kk + 1) * NCOLS];            // B[kk+2h+1][n0+r]
        acc = __builtin_amdgcn_wmma_f32_16x16x4_f32(
            /*neg_a=*/false, a, /*neg_b=*/false, b,
            /*c_mod=*/(short)0, acc, /*reuse_a=*/false, /*reuse_b=*/false);
    }

#pragma unroll
    for (int v = 0; v < 8; ++v) {
        int row = m0 + v + 8 * half;
        if (row < M) C[(long)row * NCOLS + n0 + r] = acc[v];
    }
}

// ---------------------------------------------------------------------------
// Scatter phase:  out[d] = sum_{e: dst=d} h[src_e] * dinv[src]*dinv[dst] + b
// Self loop handled in init (norm = dinv[i]^2), then edge atomics.
// ---------------------------------------------------------------------------
template <int C>
__global__ void scatter_init_kernel(const float* __restrict__ h,
                                    const float* __restrict__ dinv,
                                    const float* __restrict__ bias,
                                    float* __restrict__ out, int n) {
    int i = blockIdx.x * blockDim.x + threadIdx.x;
    if (i >= n * C) return;
    int node = i / C;
    int c    = i - node * C;
    float w = dinv[node];
    out[i] = h[i] * w * w + bias[c];
}

template <int C>
__global__ void scatter_edges_kernel(const float* __restrict__ h,
                                     const float* __restrict__ dinv,
                                     const long long* __restrict__ src,
                                     const long long* __restrict__ dst,
                                     int E, float* __restrict__ out) {
    int e = blockIdx.x * (blockDim.x >> 5) + (threadIdx.x >> 5);
    if (e >= E) return;
    int lane = threadIdx.x & 31;
    int s = (int)src[e];
    int d = (int)dst[e];
    float w = dinv[s] * dinv[d];
    const float* __restrict__ hs = h + (long)s * C;
    float* __restrict__ od = out + (long)d * C;
#pragma unroll
    for (int c = lane; c < C; c += 32) {
        atomicAdd(&od[c], hs[c] * w);
    }
}

__global__ void relu_kernel(float* __restrict__ x, long n) {
    long i = (long)blockIdx.x * blockDim.x + threadIdx.x;
    if (i < n) x[i] = fmaxf(x[i], 0.0f);
}

// ---------------------------------------------------------------------------
// logits[e] = dot(z[a], z[b]) over 64 channels; edges = concat(pos, neg)
// ---------------------------------------------------------------------------
__global__ void logits_kernel(const float* __restrict__ z,
                              const long long* __restrict__ pos,
                              const long long* __restrict__ neg,
                              int Ep, int En, float* __restrict__ out) {
    int e = blockIdx.x * blockDim.x + threadIdx.x;
    if (e >= Ep + En) return;
    long a, b;
    if (e < Ep) { a = pos[e];      b = pos[e + Ep]; }
    else { int f = e - Ep; a = neg[f]; b = neg[f + En]; }
    const float4* __restrict__ za = (const float4*)(z + a * 64);
    const float4* __restrict__ zb = (const float4*)(z + b * 64);
    float acc = 0.0f;
#pragma unroll
    for (int i = 0; i < 16; ++i) {
        float4 p = za[i], q = zb[i];
        acc += p.x * q.x + p.y * q.y + p.z * q.z + p.w * q.w;
    }
    out[e] = acc;
}

// ---------------------------------------------------------------------------
// Orchestration
// ---------------------------------------------------------------------------
extern "C" void kernel_launch(void* const* d_in, const int* in_sizes, int n_in,
                              void* d_out, int out_size, void* d_ws, size_t ws_size,
                              hipStream_t stream) {
    constexpr int IN_C = 256, HID_C = 128, OUT_C = 64;

    const float*      x   = (const float*)d_in[0];
    const long long*  ei  = (const long long*)d_in[1];
    const long long*  pos = (const long long*)d_in[2];
    const long long*  neg = (const long long*)d_in[3];
    const float*      W1  = (const float*)d_in[4];
    const float*      b1  = (const float*)d_in[5];
    const float*      W2  = (const float*)d_in[6];
    const float*      b2  = (const float*)d_in[7];
    float*            out = (float*)d_out;

    const int n  = in_sizes[0] / IN_C;   // 50000
    const int E  = in_sizes[1] / 2;      // 800000
    const int Ep = in_sizes[2] / 2;      // 200000
    const int En = in_sizes[3] / 2;      // 200000

    const long long* src = ei;          // edge_index[0]
    const long long* dst = ei + E;      // edge_index[1]

    // Workspace carve-up (floats), 256B-aligned regions
    float* ws   = (float*)d_ws;
    long   nPad = ((long)n + 63) & ~63L;
    float* dinv = ws;                               // n
    float* h1   = dinv + nPad;                      // n*HID_C
    float* a1   = h1 + (long)n * HID_C;             // n*HID_C
    float* h2   = a1 + (long)n * HID_C;             // n*OUT_C
    float* zbuf = h2 + (long)n * OUT_C;             // n*OUT_C
    (void)ws_size; (void)n_in; (void)out_size;

    const int B = 256;
    auto cdiv = [](long a, long b) { return (int)((a + b - 1) / b); };

    // ---- degrees -> dinv ----
    deg_init_kernel<<<cdiv(n, B), B, 0, stream>>>(dinv, n);
    deg_edges_kernel<<<cdiv(E, B), B, 0, stream>>>(dst, E, dinv);
    dinv_kernel<<<cdiv(n, B), B, 0, stream>>>(dinv, n);

    // ---- layer 1: h1 = x @ W1 ----
    {
        int tiles = cdiv(n, 16) * (HID_C / 16);
        gemm_wmma_f32_kernel<IN_C, HID_C><<<cdiv(tiles, 8), B, 0, stream>>>(x, W1, h1, n);
    }
    scatter_init_kernel<HID_C><<<cdiv((long)n * HID_C, B), B, 0, stream>>>(h1, dinv, b1, a1, n);
    scatter_edges_kernel<HID_C><<<cdiv(E, 8), B, 0, stream>>>(h1, dinv, src, dst, E, a1);
    relu_kernel<<<cdiv((long)n * HID_C, B), B, 0, stream>>>(a1, (long)n * HID_C);

    // ---- layer 2: h2 = a1 @ W2 ----
    {
        int tiles = cdiv(n, 16) * (OUT_C / 16);
        gemm_wmma_f32_kernel<HID_C, OUT_C><<<cdiv(tiles, 8), B, 0, stream>>>(a1, W2, h2, n);
    }
    scatter_init_kernel<OUT_C><<<cdiv((long)n * OUT_C, B), B, 0, stream>>>(h2, dinv, b2, zbuf, n);
    scatter_edges_kernel<OUT_C><<<cdiv(E, 8), B, 0, stream>>>(h2, dinv, src, dst, E, zbuf);

    // ---- decoder ----
    logits_kernel<<<cdiv(Ep + En, B), B, 0, stream>>>(zbuf, pos, neg, Ep, En, out);
}